// FormAdjcent_61194694034278
// MI455X (gfx1250) — compile-verified
//
#include <hip/hip_runtime.h>
#include <hip/hip_bf16.h>

#define HIDDEN 768
#define EPS_F  0.0001f

typedef __attribute__((ext_vector_type(2))) float v2f;
typedef __attribute__((ext_vector_type(8))) float v8f;

// Fill adj with 1 + EPS (diagonal entries are never overwritten).
__global__ void init_adj_kernel(float* __restrict__ out, int n_elem) {
    int i = blockIdx.x * blockDim.x + threadIdx.x;
    if (i < n_elem) out[i] = 1.0f + EPS_F;
}

// One wave32 per 16-row tile of pooled_output.
// GEMV via V_WMMA_F32_16X16X4_F32 with W broadcast across all 16 N columns:
// D[m][n] = sum_k A[m][k] * W[k] for every n. Then sigmoid + scatter via pairs.
__global__ __launch_bounds__(256) void gemv_sigmoid_scatter_kernel(
    const float* __restrict__ pooled,   // P x 768
    const float* __restrict__ W,        // 768
    const float* __restrict__ bptr,     // 1
    const int*   __restrict__ pairs,    // P x 2
    float*       __restrict__ out,      // B x n x n
    int P, int ppp, int n)
{
    const int lane = threadIdx.x & 31;
    const int wave = threadIdx.x >> 5;
    const int tile = blockIdx.x * (blockDim.x >> 5) + wave;
    const int numTiles = (P + 15) >> 4;
    if (tile >= numTiles) return;          // uniform per wave: EXEC stays all-1s

    const int rowBase = tile << 4;
    const int m  = lane & 15;              // row within tile (A-matrix M)
    const int g2 = (lane >> 4) << 1;       // K sub-offset: 0 (lanes 0-15) or 2 (lanes 16-31)

    int row = rowBase + m;
    if (row >= P) row = P - 1;             // clamp (P is a multiple of 16 in practice)

    const float* arow = pooled + (size_t)row * HIDDEN + g2;
    const float* wvec = W + g2;

    v8f c0 = {};
    v8f c1 = {};

    // 192 WMMA steps of K=4; two accumulators to break the RAW chain.
    #pragma unroll 4
    for (int k = 0; k < HIDDEN; k += 8) {
        v2f a0 = *(const v2f*)(arow + k);       // A frag: K = k + g2 + {0,1}
        v2f b0 = *(const v2f*)(wvec + k);       // B frag: same K indexing, bcast over N
        c0 = __builtin_amdgcn_wmma_f32_16x16x4_f32(
                 false, a0, false, b0, (short)0, c0, false, false);

        v2f a1 = *(const v2f*)(arow + k + 4);
        v2f b1 = *(const v2f*)(wvec + k + 4);
        c1 = __builtin_amdgcn_wmma_f32_16x16x4_f32(
                 false, a1, false, b1, (short)0, c1, false, false);
    }

    v8f acc = c0 + c1;

    // D layout (16x16 f32 C/D): lane 0 -> N=0, M=0..7 in acc[0..7];
    // lane 16 -> N=0, M=8..15 in acc[0..7].
    if (lane == 0 || lane == 16) {
        const float bval = bptr[0];
        const int base = rowBase + ((lane >> 4) << 3);
        #pragma unroll
        for (int i = 0; i < 8; ++i) {
            int p = base + i;
            if (p < P) {
                int r  = pairs[2 * p];
                int cc = pairs[2 * p + 1];
                int bidx = p / ppp;
                float x = acc[i] + bval;
                float s = 1.0f / (1.0f + __expf(-x));
                out[(size_t)bidx * n * n + r * n + cc] = s + EPS_F;
            }
        }
    }
}

extern "C" void kernel_launch(void* const* d_in, const int* in_sizes, int n_in,
                              void* d_out, int out_size, void* d_ws, size_t ws_size,
                              hipStream_t stream) {
    const float* pooled = (const float*)d_in[0];
    const float* W      = (const float*)d_in[1];
    const float* b      = (const float*)d_in[2];
    const int*   pairs  = (const int*)d_in[3];
    float* out = (float*)d_out;

    const int P = in_sizes[0] / HIDDEN;            // number of pair rows
    // out_size = B*n*n, P = B*n*(n-1)  =>  out_size - P = B*n  =>  n = out_size/(B*n)
    const int Bn = out_size - P;                   // B*n
    const int n  = out_size / Bn;                  // length
    const int ppp = n * (n - 1);                   // pairs per passage

    // 1) adj = 1 + EPS everywhere
    init_adj_kernel<<<(out_size + 255) / 256, 256, 0, stream>>>(out, out_size);

    // 2) GEMV (WMMA f32 16x16x4) + sigmoid + scatter into off-diagonals
    const int numTiles = (P + 15) / 16;            // 16 rows per wave
    const int wavesPerBlock = 8;                   // 256 threads = 8 wave32
    const int blocks = (numTiles + wavesPerBlock - 1) / wavesPerBlock;
    gemv_sigmoid_scatter_kernel<<<blocks, 256, 0, stream>>>(
        pooled, W, b, pairs, out, P, ppp, n);
}